// GCNConv_84043920048429
// MI455X (gfx1250) — compile-verified
//
#include <hip/hip_runtime.h>
#include <hip/hip_bf16.h>

// ---------------------------------------------------------------------------
// GCNConv for MI455X (gfx1250), fp32 end-to-end.
//   h   = x @ W^T + b                 (V_WMMA_F32_16X16X4_F32 matrix pipe)
//   deg = 1 + count(row)              (self loops)
//   out = dinv[i]^2 * h[i]            (self-loop term, initializes d_out)
//   out[row] += dinv[row]*dinv[col]*h[col]   (coalesced float4 + HW f32 atomics)
// ---------------------------------------------------------------------------

typedef float v2f __attribute__((ext_vector_type(2)));
typedef float v8f __attribute__((ext_vector_type(8)));

#define IN_CH  256
#define OUT_CH 128

// ---------------------------------------------------------------------------
// GEMM: h[n,o] = sum_k x[n,k] * W[o,k] + b[o]
// One wave per 16x16 output tile. K loop: 64 x V_WMMA_F32_16X16X4_F32.
//
// VGPR layouts (CDNA5 ISA 7.12.2, fp32):
//   A 16x4 : lane l -> M = l&15 ; VGPR {0,1} = K = 2*(l>>4) + {0,1}
//   B 4x16 : lane l -> N = l&15 ; VGPR {0,1} = K = 2*(l>>4) + {0,1}  (B[k][n] = W[n][k])
//   C 16x16: VGPR i -> M = i + 8*(l>>4), N = l&15
// ---------------------------------------------------------------------------
__global__ void __launch_bounds__(32)
gcn_gemm_wmma(const float* __restrict__ x, const float* __restrict__ W,
              const float* __restrict__ bias, float* __restrict__ h, int n_nodes)
{
    const int lane = threadIdx.x & 31;
    const int half = lane >> 4;   // 0|1 -> K group (A/B) / M group (C)
    const int l15  = lane & 15;

    const int m0 = blockIdx.x * 16;          // node tile
    const int n0 = blockIdx.y * 16;          // output-channel tile

    int arow = m0 + l15;                     // node row this lane streams for A
    if (arow > n_nodes - 1) arow = n_nodes - 1;   // clamp: keep EXEC all-ones for WMMA
    const int brow = n0 + l15;               // W row (output channel) for B column

    const float* xa = x + (size_t)arow * IN_CH + 2 * half;
    const float* wb = W + (size_t)brow * IN_CH + 2 * half;

    v8f acc = {};
#pragma unroll 8
    for (int k = 0; k < IN_CH; k += 4) {
        v2f a = *(const v2f*)(xa + k);       // 8B aligned: (row*256 + 2*half + k) even
        v2f bm = *(const v2f*)(wb + k);
        // (neg_a, A, neg_b, B, c_mod, C, reuse_a, reuse_b)
        acc = __builtin_amdgcn_wmma_f32_16x16x4_f32(
            false, a, false, bm, (short)0, acc, false, false);
    }

    const float bv = bias[n0 + l15];
    float* hp = h + (size_t)(m0 + 8 * half) * OUT_CH + n0 + l15;

    if (m0 + 16 <= n_nodes) {
        // Fast path: full tile (always taken for N % 16 == 0), no exec-mask churn.
#pragma unroll
        for (int i = 0; i < 8; ++i)
            hp[(size_t)i * OUT_CH] = acc[i] + bv;
    } else {
        const int mbase = m0 + 8 * half;
#pragma unroll
        for (int i = 0; i < 8; ++i)
            if (mbase + i < n_nodes)
                hp[(size_t)i * OUT_CH] = acc[i] + bv;
    }
}

// deg[i] = 1.0 (self loop)
__global__ void gcn_deg_init(float* __restrict__ deg, int n)
{
    int i = blockIdx.x * blockDim.x + threadIdx.x;
    if (i < n) deg[i] = 1.0f;
}

// deg[row[e]] += 1
__global__ void gcn_deg_count(const int* __restrict__ rows, float* __restrict__ deg, int e)
{
    int i = blockIdx.x * blockDim.x + threadIdx.x;
    if (i < e) unsafeAtomicAdd(&deg[rows[i]], 1.0f);
}

// dinv[i] = deg[i]^-0.5
__global__ void gcn_rsqrt(const float* __restrict__ deg, float* __restrict__ dinv, int n)
{
    int i = blockIdx.x * blockDim.x + threadIdx.x;
    if (i < n) dinv[i] = 1.0f / sqrtf(deg[i]);
}

// out[i,:] = dinv[i]^2 * h[i,:]   (self-loop term; also initializes d_out)
__global__ void gcn_self_loop(const float* __restrict__ h, const float* __restrict__ dinv,
                              float* __restrict__ out, long total)
{
    long i = (long)blockIdx.x * blockDim.x + threadIdx.x;
    if (i >= total) return;
    const int node = (int)(i >> 7);          // /OUT_CH
    const float s = dinv[node];
    out[i] = s * s * h[i];
}

// out[row,:] += dinv[row]*dinv[col] * h[col,:]
// 8 edges / 256-thread block; 32 lanes/edge, float4/lane -> 512B coalesced rows.
__global__ void __launch_bounds__(256)
gcn_edge_scatter(const int* __restrict__ ei, const float* __restrict__ h,
                 const float* __restrict__ dinv, float* __restrict__ out, int n_edges)
{
    const int lane = threadIdx.x & 31;
    const int edge = blockIdx.x * 8 + (threadIdx.x >> 5);
    if (edge >= n_edges) return;

    const int r = ei[edge];                  // edge_index[0][e] (destination / segment id)
    const int c = ei[n_edges + edge];        // edge_index[1][e] (source)
    const float nrm = dinv[r] * dinv[c];

    const float4 hv = ((const float4*)(h + (size_t)c * OUT_CH))[lane];
    float* o = out + (size_t)r * OUT_CH + lane * 4;
    unsafeAtomicAdd(o + 0, nrm * hv.x);
    unsafeAtomicAdd(o + 1, nrm * hv.y);
    unsafeAtomicAdd(o + 2, nrm * hv.z);
    unsafeAtomicAdd(o + 3, nrm * hv.w);
}

extern "C" void kernel_launch(void* const* d_in, const int* in_sizes, int n_in,
                              void* d_out, int out_size, void* d_ws, size_t ws_size,
                              hipStream_t stream)
{
    const float* x    = (const float*)d_in[0];   // [N, 256]
    const int*   ei   = (const int*)d_in[1];     // [2, E] row-major: rows then cols
    const float* W    = (const float*)d_in[2];   // [128, 256]
    const float* bias = (const float*)d_in[3];   // [128]
    float* out = (float*)d_out;                  // [N, 128]

    const int n_nodes = in_sizes[0] / IN_CH;     // 50000
    const int n_edges = in_sizes[1] / 2;         // 1600000

    // Workspace: h [N*128] fp32, deg [N], dinv [N]  (~25.6 MB + 0.4 MB)
    float* h    = (float*)d_ws;
    float* deg  = h + (size_t)n_nodes * OUT_CH;
    float* dinv = deg + n_nodes;

    // 1) h = x @ W^T + b  (WMMA)
    {
        dim3 grid((n_nodes + 15) / 16, OUT_CH / 16);
        gcn_gemm_wmma<<<grid, 32, 0, stream>>>(x, W, bias, h, n_nodes);
    }
    // 2) degrees (self loop = 1, then count edge rows)
    gcn_deg_init<<<(n_nodes + 255) / 256, 256, 0, stream>>>(deg, n_nodes);
    gcn_deg_count<<<(n_edges + 255) / 256, 256, 0, stream>>>(ei, deg, n_edges);
    // 3) dinv = deg^-0.5
    gcn_rsqrt<<<(n_nodes + 255) / 256, 256, 0, stream>>>(deg, dinv, n_nodes);
    // 4) out = dinv^2 * h  (self-loop contribution, initializes out)
    {
        const long total = (long)n_nodes * OUT_CH;
        gcn_self_loop<<<(int)((total + 255) / 256), 256, 0, stream>>>(h, dinv, out, total);
    }
    // 5) scatter-add edge messages
    gcn_edge_scatter<<<(n_edges + 7) / 8, 256, 0, stream>>>(ei, h, dinv, out, n_edges);
}